// PixelDSNTDistanceDoubleEval_20083267076470
// MI455X (gfx1250) — compile-verified
//
#include <hip/hip_runtime.h>
#include <math.h>

typedef float f32x4 __attribute__((ext_vector_type(4)));
typedef float v2f   __attribute__((ext_vector_type(2)));
typedef float v8f   __attribute__((ext_vector_type(8)));

#define MAPS   128                 // B*C = 64*2
#define HW_N   (512 * 512)         // elements per map
#define SPLIT  16                  // blocks per map
#define T1     256                 // threads per block (8 wave32)
#define CHUNK  (HW_N / SPLIT)      // 16384 elements per block
#define ITERS  (CHUNK / (T1 * 4))  // 16 float4 iterations per thread

// ---- merge helpers -------------------------------------------------------

__device__ __forceinline__ void merge_soft(float& m, float& s, float& sx, float& sy,
                                           float om, float os, float osx, float osy) {
  float nm = fmaxf(m, om);
  float a = __expf(m - nm);
  float b = __expf(om - nm);
  s  = s  * a + os  * b;
  sx = sx * a + osx * b;
  sy = sy * a + osy * b;
  m  = nm;
}

__device__ __forceinline__ void merge_arg(float& bv, float& bi, float obv, float obi) {
  // prefer larger value; on tie prefer smaller flat index (jnp.argmax semantics)
  if (obv > bv || (obv == bv && obi < bi)) { bv = obv; bi = obi; }
}

// ---- kernel 1: streaming per-chunk partials (bandwidth-bound) ------------

__global__ __launch_bounds__(T1) void dsnt_partial_kernel(
    const float* __restrict__ inp, const float* __restrict__ tgt,
    float* __restrict__ ws) {
  const int tid = threadIdx.x;
  const unsigned bid  = blockIdx.x;
  const unsigned map  = bid >> 4;            // bid / SPLIT
  const unsigned part = bid & (SPLIT - 1);
  const size_t   mapBase   = (size_t)map * HW_N;
  const unsigned chunkBase = part * CHUNK;   // element offset within map

  const f32x4* in4 = (const f32x4*)(inp + mapBase);
  const f32x4* tg4 = (const f32x4*)(tgt + mapBase);

  // online softmax state (pixel-unit weights) + argmax state
  float m = -INFINITY, s = 0.f, sx = 0.f, sy = 0.f;
  float bv = -INFINITY, bi = 0.f;            // index kept in float (< 2^24, exact)

  #pragma unroll 4
  for (int i = 0; i < ITERS; ++i) {
    const unsigned e = chunkBase + (unsigned)(i * T1 + tid) * 4u; // element idx in map
    const unsigned q = e >> 2;                                    // float4 idx in map
    if (i + 4 < ITERS) {  // prefetch 4 iterations ahead (gfx1250 global_prefetch)
      __builtin_prefetch((const void*)(in4 + q + 4 * T1), 0, 1);
      __builtin_prefetch((const void*)(tg4 + q + 4 * T1), 0, 1);
    }
    // streamed once -> non-temporal b128 loads, keep caches clean
    f32x4 v = __builtin_nontemporal_load(in4 + q);
    f32x4 t = __builtin_nontemporal_load(tg4 + q);

    const float rowp = (float)((e >> 9) + 1u);   // (row+1) pixel coord
    const float colp = (float)((e & 511u) + 1u); // (col+1) of lane .x (no row cross: 4|512)

    // online softmax: rescale only when max improves (rare for N(0,1) stream)
    float lm = fmaxf(fmaxf(v.x, v.y), fmaxf(v.z, v.w));
    if (lm > m) { float sc = __expf(m - lm); s *= sc; sx *= sc; sy *= sc; m = lm; }
    float e0 = __expf(v.x - m), e1 = __expf(v.y - m);
    float e2 = __expf(v.z - m), e3 = __expf(v.w - m);
    float es = e0 + e1 + e2 + e3;
    s  += es;
    sx += e0 * colp + e1 * (colp + 1.f) + e2 * (colp + 2.f) + e3 * (colp + 3.f);
    sy += es * rowp;

    // first-occurrence argmax over target (ascending e within thread, strict >)
    if (t.x > bv) { bv = t.x; bi = (float)(e + 0u); }
    if (t.y > bv) { bv = t.y; bi = (float)(e + 1u); }
    if (t.z > bv) { bv = t.z; bi = (float)(e + 2u); }
    if (t.w > bv) { bv = t.w; bi = (float)(e + 3u); }
  }

  // wave32 shuffle reduction
  for (int off = 16; off > 0; off >>= 1) {
    float om  = __shfl_down(m,  off, 32);
    float os  = __shfl_down(s,  off, 32);
    float osx = __shfl_down(sx, off, 32);
    float osy = __shfl_down(sy, off, 32);
    float obv = __shfl_down(bv, off, 32);
    float obi = __shfl_down(bi, off, 32);
    merge_soft(m, s, sx, sy, om, os, osx, osy);
    merge_arg(bv, bi, obv, obi);
  }

  __shared__ float red[T1 / 32][6];
  const int wid = tid >> 5, lane = tid & 31;
  if (lane == 0) {
    red[wid][0] = m;  red[wid][1] = s;  red[wid][2] = sx;
    red[wid][3] = sy; red[wid][4] = bv; red[wid][5] = bi;
  }
  __syncthreads();
  if (tid == 0) {
    for (int w = 1; w < T1 / 32; ++w) {
      merge_soft(m, s, sx, sy, red[w][0], red[w][1], red[w][2], red[w][3]);
      merge_arg(bv, bi, red[w][4], red[w][5]);
    }
    float* p = ws + (size_t)bid * 8;
    p[0] = m; p[1] = s; p[2] = sx; p[3] = sy; p[4] = bv; p[5] = bi;
  }
}

// ---- f32 WMMA 32-lane sum: A = per-lane partials, B = ones ---------------
// A(16x4): lane L<16 holds (M=L,K=0),(M=L,K=1); lane L>=16 holds (M=L-16,K=2),(K=3).
// With B=1: D[m,n] = sum_k A[m,k] = p[m] + p[m+16]. D vgpr j: lanes0-15 = row j,
// lanes16-31 = row j+8. Sum c[0..7] then shfl_xor(16) -> total over all 32 lanes.
__device__ __forceinline__ float wmma_sum32(float p) {
  v2f a;  a.x = p;   a.y = 0.f;
  v2f bo; bo.x = 1.f; bo.y = 1.f;
  v8f c = {};
  c = __builtin_amdgcn_wmma_f32_16x16x4_f32(false, a, false, bo, (short)0, c,
                                            false, false);
  float r = c[0] + c[1] + c[2] + c[3] + c[4] + c[5] + c[6] + c[7];
  r += __shfl_xor(r, 16, 32);
  return r;
}

// ---- kernel 2: merge partials, per-batch math, final sums ----------------
// One wave32. Lane L owns maps 4L..4L+3 == batches b=2L and b=2L+1 (both keypoints).

__global__ __launch_bounds__(32) void dsnt_final_kernel(
    const float* __restrict__ ws, float* __restrict__ out) {
  const int L = threadIdx.x;
  float ed[4], px[4], py[4], tx[4], ty[4];
  float p_si = 0.f, p_ss = 0.f, p_sx = 0.f, p_sy = 0.f;

  for (int j = 0; j < 4; ++j) {
    const int map = L * 4 + j;
    float m = -INFINITY, s = 0.f, sx = 0.f, sy = 0.f;
    float bv = -INFINITY, bi = 0.f;
    for (int pI = 0; pI < SPLIT; ++pI) {
      const float* pp = ws + (size_t)(map * SPLIT + pI) * 8;
      merge_soft(m, s, sx, sy, pp[0], pp[1], pp[2], pp[3]);
      merge_arg(bv, bi, pp[4], pp[5]);
    }
    const float predx = sx / s;               // already pixel units
    const float predy = sy / s;
    const int   ib    = (int)bi;
    const float truex = (float)((ib & 511) + 1);
    const float truey = (float)((ib >> 9) + 1);
    const float xd = truex - predx;
    const float yd = truey - predy;
    ed[j] = sqrtf(xd * xd + yd * yd);
    px[j] = predx; py[j] = predy; tx[j] = truex; ty[j] = truey;
    p_sx += fabsf(xd);
    p_sy += fabsf(yd);
    if ((map & 1) == 0) p_si += ed[j]; else p_ss += ed[j];
  }

  // per-batch outputs (b0 = 2L from maps j=0,1; b1 = 2L+1 from maps j=2,3)
  const float tot0 = ed[0] + ed[1];
  const float tot1 = ed[2] + ed[3];
  float vpx = px[0] - px[1], vpy = py[0] - py[1];
  float vtx = tx[0] - tx[1], vty = ty[0] - ty[1];
  const float d0 = fabsf(sqrtf(vpx * vpx + vpy * vpy) - sqrtf(vtx * vtx + vty * vty));
  vpx = px[2] - px[3]; vpy = py[2] - py[3];
  vtx = tx[2] - tx[3]; vty = ty[2] - ty[3];
  const float d1 = fabsf(sqrtf(vpx * vpx + vpy * vpy) - sqrtf(vtx * vtx + vty * vty));
  const float p_sd = d0 + d1;

  // out layout: [0]=s_i [1]=s_s [2]=s_i+s_s [3]=s_diam [4..67]=tot_list
  //             [68..131]=diam_list [132]=s_x [133]=s_y [134]=B
  out[4 + 2 * L]      = tot0;
  out[4 + 2 * L + 1]  = tot1;
  out[68 + 2 * L]     = d0;
  out[68 + 2 * L + 1] = d1;

  // all 32 lanes active here: EXEC all-ones as WMMA requires
  const float s_i = wmma_sum32(p_si);
  const float s_s = wmma_sum32(p_ss);
  const float s_d = wmma_sum32(p_sd);
  const float s_x = wmma_sum32(p_sx);
  const float s_y = wmma_sum32(p_sy);
  if (L == 0) {
    out[0] = s_i; out[1] = s_s; out[2] = s_i + s_s; out[3] = s_d;
    out[132] = s_x; out[133] = s_y; out[134] = 64.0f;
  }
}

// ---- host entry ----------------------------------------------------------

extern "C" void kernel_launch(void* const* d_in, const int* in_sizes, int n_in,
                              void* d_out, int out_size, void* d_ws, size_t ws_size,
                              hipStream_t stream) {
  (void)in_sizes; (void)n_in; (void)out_size; (void)ws_size;
  const float* inp = (const float*)d_in[0];
  const float* tgt = (const float*)d_in[1];
  float* out = (float*)d_out;
  float* ws  = (float*)d_ws;   // needs MAPS*SPLIT*8 floats = 64 KiB

  dsnt_partial_kernel<<<dim3(MAPS * SPLIT), dim3(T1), 0, stream>>>(inp, tgt, ws);
  dsnt_final_kernel<<<dim3(1), dim3(32), 0, stream>>>(ws, out);
}